// SpatialAttention_62921270886792
// MI455X (gfx1250) — compile-verified
//
#include <hip/hip_runtime.h>
#include <hip/hip_bf16.h>
#include <stdint.h>

// Problem dims (fixed by the reference setup_inputs()).
#define B_  32
#define C_  128
#define H_  128
#define W_  128
#define HW_ (H_ * W_)

// Tile geometry for the conv+scale kernel.
#define TS   32              // spatial tile (TS x TS)
#define HALO 3               // 7x7 conv, pad 3
#define TD   (TS + 2 * HALO) // 38 loaded rows/cols
#define TP   40              // padded LDS row pitch (floats)

#define MLP  8               // explicit loads-in-flight per thread

// Native clang vector type: required by __builtin_nontemporal_*.
typedef float v4f __attribute__((ext_vector_type(4)));

// ---- CDNA5 async global->LDS path (guarded; falls back to sync copy) ----
#if defined(__gfx1250__) && __has_builtin(__builtin_amdgcn_global_load_async_to_lds_b32)
#define USE_ASYNC_LDS 1
#else
#define USE_ASYNC_LDS 0
#endif

typedef __attribute__((address_space(1))) int glb_i32;
typedef __attribute__((address_space(3))) int lds_i32;

__device__ __forceinline__ void async_wait_zero() {
#if defined(__gfx1250__)
#if __has_builtin(__builtin_amdgcn_s_wait_asynccnt)
  __builtin_amdgcn_s_wait_asynccnt(0);
#else
  asm volatile("s_wait_asynccnt 0" ::: "memory");
#endif
#endif
}

// ---------------------------------------------------------------------------
// Pass 1: channel max + mean reduction.
//   x   : [B, C, H, W]
//   map : [B, 2, H, W]  (plane 0 = avg, plane 1 = max; matches concat order)
// Each thread owns 4 consecutive pixels (one v4f) and walks 128 channels in
// explicit batches of MLP loads so the scheduler keeps 8 NT b128 loads in
// flight per thread (131K threads * 128 B ≈ 16 MB in flight, matching the
// 23.3 TB/s x ~600 ns latency-bandwidth product).
// ---------------------------------------------------------------------------
__global__ __launch_bounds__(256) void k_reduce(const float* __restrict__ x,
                                                float* __restrict__ map) {
  const int i    = blockIdx.x * 256 + threadIdx.x; // v4f-group id
  const int base = i * 4;                          // pixel index within [0, B*HW)
  const int b    = base / HW_;
  const int rem  = base - b * HW_;

  const v4f* xp  = (const v4f*)(x + (size_t)b * C_ * HW_ + rem);
  const int  cs4 = HW_ / 4; // channel stride in v4f units

  const float ninf = -__builtin_inff();
  v4f mx = {ninf, ninf, ninf, ninf};
  v4f sm = {0.0f, 0.0f, 0.0f, 0.0f};

  for (int c0 = 0; c0 < C_; c0 += MLP) {
    v4f tb[MLP];
    // Phase 1: issue MLP independent NT loads back-to-back.
#pragma unroll
    for (int u = 0; u < MLP; ++u) {
      tb[u] = __builtin_nontemporal_load(xp + (c0 + u) * cs4);
    }
    // Speculative prefetch two batches ahead -> global_prefetch_b8.
    if (c0 + 2 * MLP < C_) {
      __builtin_prefetch((const void*)(xp + (c0 + 2 * MLP) * cs4), 0, 1);
    }
    // Phase 2: consume (waits retire loads oldest-first).
#pragma unroll
    for (int u = 0; u < MLP; ++u) {
      v4f t = tb[u];
      mx.x = fmaxf(mx.x, t.x); mx.y = fmaxf(mx.y, t.y);
      mx.z = fmaxf(mx.z, t.z); mx.w = fmaxf(mx.w, t.w);
      sm += t;
    }
  }

  const float inv = 1.0f / (float)C_;
  v4f av = sm * inv;

  // Regular-temporal stores: keep the 4 MB map L2-resident for pass 2.
  v4f* avgp = (v4f*)(map + (size_t)b * 2 * HW_ + rem);
  v4f* maxp = (v4f*)(map + (size_t)b * 2 * HW_ + HW_ + rem);
  *avgp = av;
  *maxp = mx;
}

// ---------------------------------------------------------------------------
// Pass 2: 7x7 conv on the 2-channel map (LDS-tiled via async-to-LDS DMA),
// sigmoid, then stream all 128 channels of x * s -> out.
// Grid: B * 16 blocks; block = one 32x32 tile; thread = 4 consecutive pixels.
// ---------------------------------------------------------------------------
__global__ __launch_bounds__(256) void k_conv_scale(const float* __restrict__ x,
                                                    const float* __restrict__ wt,
                                                    const float* __restrict__ map,
                                                    float* __restrict__ out) {
  __shared__ float smap[2 * TD * TP];

  const int blk = blockIdx.x;
  const int b   = blk >> 4;
  const int t   = blk & 15;
  const int th  = (t >> 2) * TS;
  const int tw  = (t & 3) * TS;
  const int tid = threadIdx.x;

  const float* mbase = map + (size_t)b * 2 * HW_;

  // Fill 2 x 38x38 halo tiles. In-bounds lanes use async global->LDS DMA
  // (ASYNCcnt-tracked, no VGPR round-trip); out-of-bounds lanes write the
  // conv's zero padding directly.
  for (int idx = tid; idx < 2 * TD * TD; idx += 256) {
    const int ic = idx / (TD * TD);
    const int r  = idx - ic * (TD * TD);
    const int lh = r / TD;
    const int lw = r - lh * TD;
    const int gh = th + lh - HALO;
    const int gw = tw + lw - HALO;
    float* ldst = &smap[(ic * TD + lh) * TP + lw];
    if ((unsigned)gh < (unsigned)H_ && (unsigned)gw < (unsigned)W_) {
      const float* gsrc = mbase + (size_t)ic * HW_ + gh * W_ + gw;
#if USE_ASYNC_LDS
      __builtin_amdgcn_global_load_async_to_lds_b32(
          (glb_i32*)(uintptr_t)gsrc,   // AS(1) int*
          (lds_i32*)ldst,              // AS(3) int*
          0, 0);
#else
      *ldst = *gsrc;
#endif
    } else {
      *ldst = 0.0f;
    }
  }
#if USE_ASYNC_LDS
  async_wait_zero();
#endif
  __syncthreads();

  const int ty = tid >> 3;        // row in tile: 0..31
  const int tx = (tid & 7) * 4;   // col in tile: 0,4,...,28

  // 7x7x2 conv for 4 pixels; weights have uniform addresses -> scalar loads.
  v4f sgv;
#pragma unroll
  for (int j = 0; j < 4; ++j) {
    float acc = 0.0f;
#pragma unroll
    for (int ic = 0; ic < 2; ++ic) {
#pragma unroll
      for (int ky = 0; ky < 7; ++ky) {
#pragma unroll
        for (int kx = 0; kx < 7; ++kx) {
          acc = fmaf(wt[(ic * 7 + ky) * 7 + kx],
                     smap[(ic * TD + ty + ky) * TP + (tx + j + kx)], acc);
        }
      }
    }
    sgv[j] = 1.0f / (1.0f + __expf(-acc));
  }

  // Stream x -> out for all channels with non-temporal 16B traffic,
  // batched MLP loads first for deep load pipelining.
  const int gh = th + ty;
  const int gw = tw + tx;
  const size_t pix = (size_t)b * C_ * HW_ + (size_t)gh * W_ + gw; // c = 0
  const v4f* xp = (const v4f*)(x + pix);
  v4f*       op = (v4f*)(out + pix);
  const int cs4 = HW_ / 4;

  for (int c0 = 0; c0 < C_; c0 += MLP) {
    v4f tb[MLP];
#pragma unroll
    for (int u = 0; u < MLP; ++u) {
      tb[u] = __builtin_nontemporal_load(xp + (c0 + u) * cs4);
    }
#pragma unroll
    for (int u = 0; u < MLP; ++u) {
      v4f r = tb[u] * sgv;
      __builtin_nontemporal_store(r, op + (c0 + u) * cs4);
    }
  }
}

// ---------------------------------------------------------------------------
extern "C" void kernel_launch(void* const* d_in, const int* in_sizes, int n_in,
                              void* d_out, int out_size, void* d_ws, size_t ws_size,
                              hipStream_t stream) {
  const float* x  = (const float*)d_in[0];  // [32,128,128,128] f32
  const float* wt = (const float*)d_in[1];  // [1,2,7,7] f32
  float*       out = (float*)d_out;         // [32,128,128,128] f32
  float*       map = (float*)d_ws;          // needs B*2*H*W*4 = 4 MB scratch

  // Pass 1: 32*128*128 pixels / 4 per thread / 256 per block = 512 blocks.
  k_reduce<<<512, 256, 0, stream>>>(x, map);
  // Pass 2: 32 batches * 16 tiles = 512 blocks.
  k_conv_scale<<<512, 256, 0, stream>>>(x, wt, map, out);
}